// GCNEncoder_1726576853772
// MI455X (gfx1250) — compile-verified
//
#include <hip/hip_runtime.h>
#include <hip/hip_bf16.h>

typedef __attribute__((ext_vector_type(2))) float v2f;
typedef __attribute__((ext_vector_type(8))) float v8f;

// ---------------------------------------------------------------------------
// Elementwise helpers
// ---------------------------------------------------------------------------
__global__ void zero_f32_kernel(float* __restrict__ p, int n) {
  int i = blockIdx.x * blockDim.x + threadIdx.x;
  if (i < n) p[i] = 0.0f;
}

// deg[col[e]] += w[e]  (in-degree accumulation; self-loop +1 added later)
__global__ void deg_kernel(const long long* __restrict__ ei,
                           const float* __restrict__ w,
                           float* __restrict__ deg, int E) {
  int e = blockIdx.x * blockDim.x + threadIdx.x;
  if (e < E) {
    int c = (int)ei[(size_t)E + e];   // col = edge_index[1]
    unsafeAtomicAdd(&deg[c], w[e]);
  }
}

// dinv[i] = rsqrt(deg[i] + 1)  (self-loop weight 1.0), in place
__global__ void dinv_kernel(float* __restrict__ deg, int n) {
  int i = blockIdx.x * blockDim.x + threadIdx.x;
  if (i < n) {
    float d = deg[i] + 1.0f;
    deg[i] = (d > 0.0f) ? rsqrtf(d) : 0.0f;
  }
}

// out[i, :] = bias[:] + dinv[i]^2 * h[i, :]   (self-loop message + bias)
__global__ void selfloop_bias_kernel(const float* __restrict__ h,
                                     const float* __restrict__ dinv,
                                     const float* __restrict__ bias,
                                     float* __restrict__ out,
                                     int nNodes, int D) {
  const int dq = D >> 2;                       // float4s per row
  int idx = blockIdx.x * blockDim.x + threadIdx.x;
  int total = nNodes * dq;
  if (idx >= total) return;
  int node = idx / dq;
  int j4   = idx - node * dq;
  float s = dinv[node];
  s = s * s;
  float4 hv = ((const float4*)h)[idx];
  float4 bv = ((const float4*)bias)[j4];
  float4 o;
  o.x = bv.x + s * hv.x;
  o.y = bv.y + s * hv.y;
  o.z = bv.z + s * hv.z;
  o.w = bv.w + s * hv.w;
  ((float4*)out)[idx] = o;
}

// ---------------------------------------------------------------------------
// Edge scatter: out[col, :] += (dinv[row]*w*dinv[col]) * h[row, :]
// D/4 lanes per edge (32 for D=128 -> one wave; 16 for D=64 -> half wave).
// Lane 0 of each group loads edge metadata, broadcast via wave32 shuffles.
// ---------------------------------------------------------------------------
template <int D>
__global__ void scatter_kernel(const long long* __restrict__ ei,
                               const float* __restrict__ w,
                               const float* __restrict__ dinv,
                               const float* __restrict__ h,
                               float* __restrict__ out, int E) {
  constexpr int TPE = D / 4;                   // threads per edge
  const int j = threadIdx.x & (TPE - 1);       // lane within edge group
  const int e = blockIdx.x * (blockDim.x / TPE) + (threadIdx.x / TPE);
  if (e >= E) return;

  int rsrc = 0, csrc = 0;
  float nrm = 0.0f;
  if (j == 0) {
    rsrc = (int)ei[e];                         // row = edge_index[0]
    csrc = (int)ei[(size_t)E + e];             // col = edge_index[1]
    nrm  = dinv[rsrc] * w[e] * dinv[csrc];
  }
  const int   r = __shfl(rsrc, 0, TPE);
  const int   c = __shfl(csrc, 0, TPE);
  const float n = __shfl(nrm,  0, TPE);

  float4 hv = ((const float4*)(h + (size_t)r * D))[j];
  float* op = out + (size_t)c * D + 4 * j;
  unsafeAtomicAdd(op + 0, n * hv.x);
  unsafeAtomicAdd(op + 1, n * hv.y);
  unsafeAtomicAdd(op + 2, n * hv.z);
  unsafeAtomicAdd(op + 3, n * hv.w);
}

// ---------------------------------------------------------------------------
// Full-precision WMMA GEMM:  Out[nRows, NDIM] = (relu?)(A[nRows, KDIM]) * W
// Uses V_WMMA_F32_16X16X4_F32.  One wave per 16-row stripe, 8 waves/block.
// A 16x4 layout: lanes 0-15 hold K=0,1; lanes 16-31 hold K=2,3 (M = lane&15).
// B 4x16 layout: VGPR0 = row 2*half, VGPR1 = row 2*half+1, N = lane&15.
// C/D 16x16:     VGPR r -> M = r + 8*half, N = lane&15.
//
// The A-row index is CLAMPED instead of predicated: an out-of-range A row
// (M = lane&15) only contributes to the same out-of-range output row, which
// the guarded store never writes.  This keeps the K-loop branch-free (no
// per-iteration EXEC mask save/restore) and EXEC all-ones at every WMMA.
// ---------------------------------------------------------------------------
template <int KDIM, int NDIM, bool RELU>
__global__ __launch_bounds__(256) void gemm_wmma_f32(
    const float* __restrict__ A, const float* __restrict__ W,
    float* __restrict__ Out, int nRows) {
  const int lane = threadIdx.x & 31;
  const int wave = threadIdx.x >> 5;
  const int m0 = (blockIdx.x * 8 + wave) * 16;
  if (m0 >= nRows) return;                     // wave-uniform exit

  const int half = lane >> 4;
  const int l15  = lane & 15;
  const int mrow = min(m0 + l15, nRows - 1);   // clamped: branch-free loads
  constexpr int NT = NDIM / 16;

  v8f acc[NT];
#pragma unroll
  for (int t = 0; t < NT; ++t)
    acc[t] = (v8f){0.f, 0.f, 0.f, 0.f, 0.f, 0.f, 0.f, 0.f};

  const float* __restrict__ arow = A + (size_t)mrow * KDIM + 2 * half;

  for (int k = 0; k < KDIM; k += 4) {
    v2f a = *(const v2f*)(arow + k);
    if (RELU) { a.x = fmaxf(a.x, 0.f); a.y = fmaxf(a.y, 0.f); }

    const float* __restrict__ wrow = W + (size_t)(k + 2 * half) * NDIM + l15;
#pragma unroll
    for (int t = 0; t < NT; ++t) {
      v2f b;
      b.x = wrow[t * 16];
      b.y = wrow[t * 16 + NDIM];
      acc[t] = __builtin_amdgcn_wmma_f32_16x16x4_f32(
          false, a, false, b, (short)0, acc[t], false, false);
    }
  }

#pragma unroll
  for (int t = 0; t < NT; ++t) {
    const int col = t * 16 + l15;
#pragma unroll
    for (int r = 0; r < 8; ++r) {
      const int row = m0 + 8 * half + r;
      if (row < nRows) Out[(size_t)row * NDIM + col] = acc[t][r];
    }
  }
}

// ---------------------------------------------------------------------------
// Launch
// ---------------------------------------------------------------------------
extern "C" void kernel_launch(void* const* d_in, const int* in_sizes, int n_in,
                              void* d_out, int out_size, void* d_ws, size_t ws_size,
                              hipStream_t stream) {
  const float*      x  = (const float*)d_in[0];       // [N,128]
  const long long*  ei = (const long long*)d_in[1];   // [2,E] int64
  const float*      ew = (const float*)d_in[2];       // [E]
  const float*      W1 = (const float*)d_in[3];       // [128,128]
  const float*      b1 = (const float*)d_in[4];       // [128]
  const float*      W2 = (const float*)d_in[5];       // [128,64]
  const float*      b2 = (const float*)d_in[6];       // [64]
  float*            out = (float*)d_out;              // [N,64]

  const int N = in_sizes[0] / 128;
  const int E = in_sizes[2];

  float* dinv = (float*)d_ws;                         // [N]
  float* h1   = dinv + N;                             // [N,128]
  float* agg1 = h1 + (size_t)N * 128;                 // [N,128]
  float* h2   = h1;                                   // reuse h1 buffer for h2 [N,64]

  const int B = 256;
  const int gemmGrid = (((N + 15) / 16) + 7) / 8;

  // degree / normalization
  zero_f32_kernel<<<(N + B - 1) / B, B, 0, stream>>>(dinv, N);
  deg_kernel<<<(E + B - 1) / B, B, 0, stream>>>(ei, ew, dinv, E);
  dinv_kernel<<<(N + B - 1) / B, B, 0, stream>>>(dinv, N);

  // layer 1: h1 = x @ W1 ; agg1 = b1 + selfloop + edge scatter
  gemm_wmma_f32<128, 128, false><<<gemmGrid, B, 0, stream>>>(x, W1, h1, N);
  {
    int total = N * (128 / 4);
    selfloop_bias_kernel<<<(total + B - 1) / B, B, 0, stream>>>(h1, dinv, b1, agg1, N, 128);
  }
  {
    int groupsPerBlock = B / 32;                      // 32 lanes per edge
    scatter_kernel<128><<<(E + groupsPerBlock - 1) / groupsPerBlock, B, 0, stream>>>(
        ei, ew, dinv, h1, agg1, E);
  }

  // layer 2: h2 = relu(agg1) @ W2 ; out = b2 + selfloop + edge scatter
  gemm_wmma_f32<128, 64, true><<<gemmGrid, B, 0, stream>>>(agg1, W2, h2, N);
  {
    int total = N * (64 / 4);
    selfloop_bias_kernel<<<(total + B - 1) / B, B, 0, stream>>>(h2, dinv, b2, out, N, 64);
  }
  {
    int groupsPerBlock = B / 16;                      // 16 lanes per edge
    scatter_kernel<64><<<(E + groupsPerBlock - 1) / groupsPerBlock, B, 0, stream>>>(
        ei, ew, dinv, h2, out, E);
  }
}